// SampleAll_88450556494641
// MI455X (gfx1250) — compile-verified
//
#include <hip/hip_runtime.h>

typedef __attribute__((ext_vector_type(2))) float v2f;
typedef __attribute__((ext_vector_type(8))) float v8f;

#define EDIM 128
#define WAVES_PER_BLOCK 4
#define EDGES_PER_WAVE 16
#define EDGES_PER_BLOCK (WAVES_PER_BLOCK * EDGES_PER_WAVE)

// dots[e] = sum_j (tokeys @ emb[si[e]])_j * relations[pi[e]]_j * (toqueries @ emb[oi[e]])_j / sqrt(128)
//
// One wave32 handles 16 edges:
//   - gather 16 s-rows and 16 o-rows (512B each) into LDS with coalesced b128 loads
//   - semb tile (16x128) = S @ tokeys^T via v_wmma_f32_16x16x4_f32 (8 N-tiles x 32 K-chunks)
//   - oemb computed per N-tile and folded immediately with the relation row
//   - cross-lane reduction over N with shfl_xor inside each 16-lane half
__global__ __launch_bounds__(WAVES_PER_BLOCK * 32)
void rel_dot_wmma_kernel(const float* __restrict__ emb,
                         const float* __restrict__ relations,
                         const float* __restrict__ tokeys,
                         const float* __restrict__ toqueries,
                         const int* __restrict__ si,
                         const int* __restrict__ pi,
                         const int* __restrict__ oi,
                         float* __restrict__ out,
                         int E)
{
    extern __shared__ float smem[];                       // 2 * 16 * 128 floats per wave
    __shared__ int sPi[WAVES_PER_BLOCK][EDGES_PER_WAVE];

    const int lane = threadIdx.x & 31;
    const int wave = threadIdx.x >> 5;
    const int base = (blockIdx.x * WAVES_PER_BLOCK + wave) * EDGES_PER_WAVE;

    float* sS = smem + wave * (2 * EDGES_PER_WAVE * EDIM);
    float* sO = sS + EDGES_PER_WAVE * EDIM;

    // ---- coalesced gather: one global_load_b128 per lane == full 512B row per issue ----
    #pragma unroll 4
    for (int r = 0; r < EDGES_PER_WAVE; ++r) {
        const int e   = min(base + r, E - 1);
        const int srw = si[e];
        const int orw = oi[e];
        const float4 vs = *(const float4*)(emb + (size_t)srw * EDIM + lane * 4);
        const float4 vo = *(const float4*)(emb + (size_t)orw * EDIM + lane * 4);
        *(float4*)(sS + r * EDIM + lane * 4) = vs;
        *(float4*)(sO + r * EDIM + lane * 4) = vo;
    }
    if (lane < EDGES_PER_WAVE) sPi[wave][lane] = pi[min(base + lane, E - 1)];
    __syncthreads();

    const int half = lane >> 4;   // 0: lanes 0-15, 1: lanes 16-31
    const int lr   = lane & 15;

    // ---- semb = S @ tokeys^T : 8 N-tiles of 16x16, K chained in chunks of 4 ----
    // A layout (16x4 f32): VGPR v, lane L -> M = L%16, K = kc*4 + v + 2*(L/16)
    // B layout (4x16 f32): VGPR v, lane L -> N = L%16, K = kc*4 + v + 2*(L/16); B[k][n] = tokeys[n][k]
    v8f accS[8];
    #pragma unroll
    for (int t = 0; t < 8; ++t) accS[t] = (v8f)0.0f;

    for (int kc = 0; kc < EDIM / 4; ++kc) {
        const int col = kc * 4 + 2 * half;
        const v2f a = *(const v2f*)(sS + lr * EDIM + col);    // ds_load_b64
        #pragma unroll
        for (int t = 0; t < 8; ++t) {
            const v2f b = *(const v2f*)(tokeys + (size_t)(t * 16 + lr) * EDIM + col);
            accS[t] = __builtin_amdgcn_wmma_f32_16x16x4_f32(
                false, a, false, b, (short)0, accS[t], false, false);
        }
    }

    // ---- oemb per N-tile, folded immediately: part[v] += semb * pemb * oemb ----
    // C/D layout: VGPR v, lane L -> M = v + 8*half, N = t*16 + (L%16)
    v8f part = (v8f)0.0f;
    for (int t = 0; t < 8; ++t) {
        v8f accO = (v8f)0.0f;
        for (int kc = 0; kc < EDIM / 4; ++kc) {
            const int col = kc * 4 + 2 * half;
            const v2f a = *(const v2f*)(sO + lr * EDIM + col);
            const v2f b = *(const v2f*)(toqueries + (size_t)(t * 16 + lr) * EDIM + col);
            accO = __builtin_amdgcn_wmma_f32_16x16x4_f32(
                false, a, false, b, (short)0, accO, false, false);
        }
        #pragma unroll
        for (int v = 0; v < 8; ++v) {
            const int prow = sPi[wave][v + 8 * half];
            const float p  = relations[(size_t)prow * EDIM + t * 16 + lr];
            part[v] += accS[t][v] * accO[v] * p;
        }
    }

    // ---- reduce over N (16 lanes within each half), write 16 dots ----
    const float scale = 0.08838834764831845f;  // 1/sqrt(128)
    #pragma unroll
    for (int v = 0; v < 8; ++v) {
        float s = part[v];
        s += __shfl_xor(s, 1, 32);
        s += __shfl_xor(s, 2, 32);
        s += __shfl_xor(s, 4, 32);
        s += __shfl_xor(s, 8, 32);
        if (lr == 0) {
            const int e = base + v + 8 * half;   // M = v + 8*half
            if (e < E) out[e] = s * scale;
        }
    }
}

extern "C" void kernel_launch(void* const* d_in, const int* in_sizes, int n_in,
                              void* d_out, int out_size, void* d_ws, size_t ws_size,
                              hipStream_t stream) {
    const float* emb       = (const float*)d_in[0];
    const float* relations = (const float*)d_in[1];
    const float* tokeys    = (const float*)d_in[2];
    const float* toqueries = (const float*)d_in[3];
    const int*   si        = (const int*)d_in[4];
    const int*   pi        = (const int*)d_in[5];
    const int*   oi        = (const int*)d_in[6];
    float*       out       = (float*)d_out;

    const int E = in_sizes[4];
    const int blocks = (E + EDGES_PER_BLOCK - 1) / EDGES_PER_BLOCK;
    const size_t shmem = (size_t)WAVES_PER_BLOCK * 2 * EDGES_PER_WAVE * EDIM * sizeof(float);

    rel_dot_wmma_kernel<<<blocks, WAVES_PER_BLOCK * 32, shmem, stream>>>(
        emb, relations, tokeys, toqueries, si, pi, oi, out, E);
}